// EyesMouthLoss_38448547234842
// MI455X (gfx1250) — compile-verified
//
#include <hip/hip_runtime.h>
#include <hip/hip_bf16.h>

#define RADIUS 15.0f
#define WMULT  300.0f
// B=16, C=3, H=512, W=512. Tiles: 64 wide x 16 tall -> 8*32=256 tiles/image, 4096 blocks.
// 256 threads/block, each thread = one 4-pixel quad x 3 channels.
#define NBLK 4096
#define NTOT 12582912.0f   // 16*3*512*512

typedef float v2f __attribute__((ext_vector_type(2)));
typedef float v8f __attribute__((ext_vector_type(8)));

__global__ __launch_bounds__(256) void eml_main_kernel(
    const float* __restrict__ pred, const float* __restrict__ target,
    const int* __restrict__ lms, float* __restrict__ partials)
{
    __shared__ float lmx[32], lmy[32], lmeye[32];
    __shared__ int   nLmS;
    __shared__ float wsum[8];

    const int blk  = blockIdx.x;
    const int b    = blk >> 8;        // 256 tiles per image
    const int tile = blk & 255;
    const int tx   = tile & 7;        // 8 tiles across, 64 px each
    const int ty   = tile >> 3;       // 32 tiles down, 16 px each
    const int x0   = tx << 6;
    const int y0   = ty << 4;
    const int t    = threadIdx.x;

    // --- landmark culling: wave0 compacts landmarks hitting tile AABB (+radius) ---
    if (t < 32) {
        const int base = (b * 68 + 36 + t) * 2;
        const float cx = fminf(fmaxf((float)lms[base + 0], 0.0f), 511.0f);
        const float cy = fminf(fmaxf((float)lms[base + 1], 0.0f), 511.0f);
        const bool pass = (cx >= (float)x0 - RADIUS) && (cx <= (float)(x0 + 63) + RADIUS) &&
                          (cy >= (float)y0 - RADIUS) && (cy <= (float)(y0 + 15) + RADIUS);
        const unsigned mask = (unsigned)__ballot(pass);   // wave32: low 32 bits
        if (pass) {
            const int pos = __popc(mask & ((1u << t) - 1u));
            lmx[pos]   = cx;
            lmy[pos]   = cy;
            lmeye[pos] = (t < 12) ? 1.0f : 0.0f;          // 36..47 eye, 48..67 mouth
        }
        if (t == 0) nLmS = __popc(mask);
    }
    __syncthreads();

    const int qx = t & 15, qy = t >> 4;
    const int x  = x0 + (qx << 2);
    const int y  = y0 + qy;
    const size_t imgOff = (size_t)b * 3u * 262144u + (size_t)y * 512u + (size_t)x;

    // start the streams early; mask math overlaps the prefetch
    __builtin_prefetch(pred + imgOff, 0, 0);
    __builtin_prefetch(target + imgOff, 0, 0);

    // --- priority weights for the 4 pixels of this quad ---
    float w0 = 1.0f, w1 = 1.0f, w2 = 1.0f, w3 = 1.0f;
    const int n = nLmS;                                   // block-uniform
    if (n > 0) {
        const float BIG = 1e18f;
        float e0 = BIG, e1 = BIG, e2 = BIG, e3 = BIG;
        float m0 = BIG, m1 = BIG, m2 = BIG, m3 = BIG;
        const float yf = (float)y, xf = (float)x;
        for (int j = 0; j < n; ++j) {
            const float cx = lmx[j], cy = lmy[j];
            const float dy  = yf - cy;
            const float dy2 = dy * dy;
            const float dx0 = xf - cx, dx1 = dx0 + 1.0f, dx2 = dx0 + 2.0f, dx3 = dx0 + 3.0f;
            const float d0 = fmaf(dx0, dx0, dy2);
            const float d1 = fmaf(dx1, dx1, dy2);
            const float d2 = fmaf(dx2, dx2, dy2);
            const float d3 = fmaf(dx3, dx3, dy2);
            if (lmeye[j] > 0.5f) {                        // block-uniform branch
                e0 = fminf(e0, d0); e1 = fminf(e1, d1);
                e2 = fminf(e2, d2); e3 = fminf(e3, d3);
            } else {
                m0 = fminf(m0, d0); m1 = fminf(m1, d1);
                m2 = fminf(m2, d2); m3 = fminf(m3, d3);
            }
        }
        #define EML_PRIO(de, dm)                                                        \
            fminf(fminf(fmaxf(fmaf(sqrtf(de), -(1.0f/RADIUS), 1.0f), 0.0f), 1.0f) +     \
                  fminf(fmaxf(fmaf(sqrtf(dm), -(1.0f/RADIUS), 1.0f), 0.0f), 1.0f), 1.0f)
        w0 = fmaf(EML_PRIO(e0, m0), WMULT - 1.0f, 1.0f);
        w1 = fmaf(EML_PRIO(e1, m1), WMULT - 1.0f, 1.0f);
        w2 = fmaf(EML_PRIO(e2, m2), WMULT - 1.0f, 1.0f);
        w3 = fmaf(EML_PRIO(e3, m3), WMULT - 1.0f, 1.0f);
        #undef EML_PRIO
    }

    // --- streaming weighted L1, 3 channels share the quad's weights ---
    float sum = 0.0f;
    #pragma unroll
    for (int c = 0; c < 3; ++c) {
        const float4 p = *reinterpret_cast<const float4*>(pred   + imgOff + (size_t)c * 262144u);
        const float4 q = *reinterpret_cast<const float4*>(target + imgOff + (size_t)c * 262144u);
        sum = fmaf(fabsf(p.x - q.x), w0, sum);
        sum = fmaf(fabsf(p.y - q.y), w1, sum);
        sum = fmaf(fabsf(p.z - q.z), w2, sum);
        sum = fmaf(fabsf(p.w - q.w), w3, sum);
    }

    // --- deterministic block reduction: wave32 shfl tree + LDS across 8 waves ---
    #pragma unroll
    for (int off = 16; off > 0; off >>= 1)
        sum += __shfl_down(sum, off, 32);
    if ((t & 31) == 0) wsum[t >> 5] = sum;
    __syncthreads();
    if (t < 8) {
        float s = wsum[t];
        s += __shfl_down(s, 4, 32);
        s += __shfl_down(s, 2, 32);
        s += __shfl_down(s, 1, 32);
        if (t == 0) partials[blk] = s;
    }
}

// Final reduce: single wave32, EXEC all-ones. Each V_WMMA_F32_16X16X4_F32 sums a
// 16x4 f32 A fragment (64 partials in 2 VGPRs) against all-ones B into the f32
// accumulator C (exact f32 FMA). Column N=0 of C sits at lane 0 (M=0..7) and
// lane 16 (M=8..15) across the 8 C VGPRs.
__global__ __launch_bounds__(32) void eml_final_kernel(
    const float* __restrict__ partials, float* __restrict__ out)
{
    const int lane = threadIdx.x;
    v8f acc = {};
    v2f ones; ones.x = 1.0f; ones.y = 1.0f;
    #pragma unroll 4
    for (int i = 0; i < NBLK / 64; ++i) {
        v2f a;
        a.x = partials[i * 64 + lane];
        a.y = partials[i * 64 + 32 + lane];
        acc = __builtin_amdgcn_wmma_f32_16x16x4_f32(
            /*neg_a=*/false, a, /*neg_b=*/false, ones,
            /*c_mod=*/(short)0, acc, /*reuse_a=*/false, /*reuse_b=*/false);
    }
    float s = acc[0] + acc[1] + acc[2] + acc[3] + acc[4] + acc[5] + acc[6] + acc[7];
    const float total = __shfl(s, 0, 32) + __shfl(s, 16, 32);
    if (lane == 0) out[0] = total * (1.0f / NTOT);
}

extern "C" void kernel_launch(void* const* d_in, const int* in_sizes, int n_in,
                              void* d_out, int out_size, void* d_ws, size_t ws_size,
                              hipStream_t stream) {
    const float* pred    = (const float*)d_in[0];
    const float* target  = (const float*)d_in[1];
    const int*   lms     = (const int*)d_in[2];
    float*       parts   = (float*)d_ws;     // NBLK floats = 16 KB
    (void)in_sizes; (void)n_in; (void)out_size; (void)ws_size;

    eml_main_kernel<<<NBLK, 256, 0, stream>>>(pred, target, lms, parts);
    eml_final_kernel<<<1, 32, 0, stream>>>(parts, (float*)d_out);
}